// PatternDetector_69226282877319
// MI455X (gfx1250) — compile-verified
//
#include <hip/hip_runtime.h>
#include <math.h>

// ---------------------------------------------------------------------------
// PatternDetector for MI455X (gfx1250, wave32).
//  - All dense GEMMs use V_WMMA_F32_16X16X4_F32 (full fp32 matrix pipe).
//  - Pairwise similarity (relu inside the reduction -> not bilinear) uses a
//    tiled LDS VALU kernel, computing only the upper triangle and mirroring.
// ---------------------------------------------------------------------------

typedef __attribute__((ext_vector_type(2))) float v2f;
typedef __attribute__((ext_vector_type(8))) float v8f;

#define B_DIM   512
#define IN_DIM  256
#define HID     512
#define PAT     10

// One wave computes one 16x16 tile of O = X(M,K) @ W(N,K)^T.
// fp32 WMMA layouts (wave32):
//   A 16x4 : lane = h*16 + idx ; v[r] = X[m0+idx][kc + 2h + r]
//   B 4x16 : v[r] = W^T[kc+2h+r][n0+idx] = W[n0+idx][kc + 2h + r]
//   D 16x16: v[r] = O[m0 + r + 8h][n0 + idx]
// MODE: 0 = (+optional per-col bias)        (similarity projections)
//       1 = bias + BatchNorm + ReLU         (layer 1)
//       2 = bias + ReLU                     (layer 2)
//       3 = bias, N guarded to Wrows (=10)  (logit layer)
template <int MODE>
__global__ __launch_bounds__(256) void k_gemm(
    const float* __restrict__ X, const float* __restrict__ W,
    float* __restrict__ O,
    const float* __restrict__ p0, const float* __restrict__ p1,
    const float* __restrict__ p2, const float* __restrict__ p3,
    const float* __restrict__ p4,
    int K, int nTilesN, int Wrows, int ldo)
{
    const int lane = threadIdx.x;                 // 0..31
    const int id   = blockIdx.x * 8 + threadIdx.y; // flat tile id (8 waves/block)
    const int mTile = id / nTilesN;
    const int nTile = id - mTile * nTilesN;
    const int m0 = mTile * 16;
    const int n0 = nTile * 16;
    const int idx = lane & 15;
    const int h   = lane >> 4;

    // Clamp W row for the N<16 tail tile (MODE 3): column n of D depends only
    // on column n of B, so duplicated rows never pollute the columns we write.
    int wr = n0 + idx;
    if (wr > Wrows - 1) wr = Wrows - 1;

    const float* xrow = X + (size_t)(m0 + idx) * K + 2 * h;
    const float* wrow = W + (size_t)wr       * K + 2 * h;

    v8f acc = {};
    for (int kc = 0; kc < K; kc += 4) {
        v2f a = *(const v2f*)(xrow + kc);
        v2f b = *(const v2f*)(wrow + kc);
        // 8 args: (neg_a, A, neg_b, B, c_mod, C, reuse_a, reuse_b)
        acc = __builtin_amdgcn_wmma_f32_16x16x4_f32(
            false, a, false, b, (short)0, acc, false, false);
    }

    const int n = n0 + idx;
    if (MODE == 3 && n >= Wrows) return;   // after WMMA: EXEC all-1s during MMA

    float scale = 1.f, add = 0.f;
    if (MODE == 1) {
        // gamma*(v+b1-mean)*rsqrt(var+eps)+beta  ==  v*scale + add
        float inv = rsqrtf(p4[n] + 1e-5f);
        scale = p1[n] * inv;
        add   = (p0[n] - p3[n]) * scale + p2[n];
    } else {
        if (p0) add = p0[n];
    }

#pragma unroll
    for (int r = 0; r < 8; ++r) {
        float v = acc[r];
        if (MODE == 1)      v = fmaxf(fmaf(v, scale, add), 0.f);
        else if (MODE == 2) v = fmaxf(v + add, 0.f);
        else                v = v + add;
        O[(size_t)(m0 + r + 8 * h) * ldo + n] = v;
    }
}

// Row-wise softmax over [512, 10]; one thread per row.
__global__ void k_softmax(const float* __restrict__ scores,
                          float* __restrict__ probs)
{
    int row = blockIdx.x * blockDim.x + threadIdx.x;
    if (row >= B_DIM) return;
    const float* s = scores + row * PAT;
    float m = s[0];
#pragma unroll
    for (int i = 1; i < PAT; ++i) m = fmaxf(m, s[i]);
    float e[PAT], sum = 0.f;
#pragma unroll
    for (int i = 0; i < PAT; ++i) { e[i] = expf(s[i] - m); sum += e[i]; }
    float inv = 1.f / sum;
#pragma unroll
    for (int i = 0; i < PAT; ++i) probs[row * PAT + i] = e[i] * inv;
}

// similarities[i][j]: sigmoid( sum_k relu(A'[i,k] + C[j,k]) * ws2[k] + bs2 )
// for i<j, mirrored to [j][i], 0 on the diagonal.  A' already has bs1 folded
// in.  One 16x16 output tile per 256-thread block; upper-triangle blocks only.
__global__ __launch_bounds__(256) void k_pairwise(
    const float* __restrict__ A, const float* __restrict__ C,
    const float* __restrict__ ws2, const float* __restrict__ bs2,
    float* __restrict__ out)
{
    const int bx = blockIdx.x, by = blockIdx.y;
    if (bx < by) return;                       // mirrored writes cover i>j

    __shared__ float As[16][66];               // pad 66: conflict-free, b64-able
    __shared__ float Cs[16][66];
    __shared__ float Ws[64];

    const int tx = threadIdx.x, ty = threadIdx.y;   // tx -> j, ty -> i
    const int t  = ty * 16 + tx;
    const int lr = t >> 4;                     // load row 0..15
    const int lc = (t & 15) << 2;              // load col (x4)

    float acc = 0.f;
    for (int kc = 0; kc < HID; kc += 64) {
        __syncthreads();
        float4 av = *(const float4*)(A + (size_t)(by * 16 + lr) * HID + kc + lc);
        float4 cv = *(const float4*)(C + (size_t)(bx * 16 + lr) * HID + kc + lc);
        As[lr][lc + 0] = av.x; As[lr][lc + 1] = av.y;
        As[lr][lc + 2] = av.z; As[lr][lc + 3] = av.w;
        Cs[lr][lc + 0] = cv.x; Cs[lr][lc + 1] = cv.y;
        Cs[lr][lc + 2] = cv.z; Cs[lr][lc + 3] = cv.w;
        if (t < 64) Ws[t] = ws2[kc + t];
        __syncthreads();
#pragma unroll
        for (int k = 0; k < 64; ++k) {
            float v = fmaxf(As[ty][k] + Cs[tx][k], 0.f);
            acc = fmaf(v, Ws[k], acc);
        }
    }

    float s = 1.f / (1.f + expf(-(acc + bs2[0])));
    const int i = by * 16 + ty;
    const int j = bx * 16 + tx;
    if (bx > by) {                             // whole tile has i<j
        out[(size_t)i * B_DIM + j] = s;
        out[(size_t)j * B_DIM + i] = s;
    } else {                                   // diagonal tile
        if (i < j) {
            out[(size_t)i * B_DIM + j] = s;
            out[(size_t)j * B_DIM + i] = s;
        } else if (i == j) {
            out[(size_t)i * B_DIM + i] = 0.f;
        }
    }
}

extern "C" void kernel_launch(void* const* d_in, const int* in_sizes, int n_in,
                              void* d_out, int out_size, void* d_ws, size_t ws_size,
                              hipStream_t stream)
{
    const float* x     = (const float*)d_in[0];
    const float* W1    = (const float*)d_in[1];
    const float* b1    = (const float*)d_in[2];
    const float* gamma = (const float*)d_in[3];
    const float* beta  = (const float*)d_in[4];
    const float* mean  = (const float*)d_in[5];
    const float* var   = (const float*)d_in[6];
    const float* W2    = (const float*)d_in[7];
    const float* b2    = (const float*)d_in[8];
    const float* W3    = (const float*)d_in[9];
    const float* b3    = (const float*)d_in[10];
    const float* Ws1a  = (const float*)d_in[11];
    const float* Ws1b  = (const float*)d_in[12];
    const float* bs1   = (const float*)d_in[13];
    const float* ws2   = (const float*)d_in[14];
    const float* bs2   = (const float*)d_in[15];

    float* out    = (float*)d_out;
    float* probs  = out;                       // [512,10]
    float* scores = out + B_DIM * PAT;         // [512,10]
    float* sims   = out + 2 * B_DIM * PAT;     // [512,512]

    float* ws = (float*)d_ws;
    float* H  = ws;                            // [512,512]
    float* H2 = H  + B_DIM * HID;              // [512,256]
    float* Am = H2 + B_DIM * (HID / 2);        // [512,512]  (= x@Ws1a^T + bs1)
    float* Cm = Am + B_DIM * HID;              // [512,512]  (= x@Ws1b^T)

    dim3 gblk(32, 8);                          // 8 waves/block, 1 tile/wave

    // H = relu(BN(x @ W1^T + b1))           tiles = 32*32 -> 128 blocks
    k_gemm<1><<<128, gblk, 0, stream>>>(x, W1, H, b1, gamma, beta, mean, var,
                                        IN_DIM, 32, HID, HID);
    // H2 = relu(H @ W2^T + b2)              tiles = 32*16 -> 64 blocks
    k_gemm<2><<<64, gblk, 0, stream>>>(H, W2, H2, b2, nullptr, nullptr, nullptr,
                                       nullptr, HID, 16, HID / 2, HID / 2);
    // scores = H2 @ W3^T + b3  (N=10)       tiles = 32*1  -> 4 blocks
    k_gemm<3><<<4, gblk, 0, stream>>>(H2, W3, scores, b3, nullptr, nullptr,
                                      nullptr, nullptr, HID / 2, 1, PAT, PAT);
    k_softmax<<<2, 256, 0, stream>>>(scores, probs);

    // A' = x @ Ws1a^T + bs1 ; C = x @ Ws1b^T
    k_gemm<0><<<128, gblk, 0, stream>>>(x, Ws1a, Am, bs1, nullptr, nullptr,
                                        nullptr, nullptr, IN_DIM, 32, HID, HID);
    k_gemm<0><<<128, gblk, 0, stream>>>(x, Ws1b, Cm, nullptr, nullptr, nullptr,
                                        nullptr, nullptr, IN_DIM, 32, HID, HID);

    k_pairwise<<<dim3(32, 32), dim3(16, 16), 0, stream>>>(Am, Cm, ws2, bs2, sims);
}